// AdderConv_new_77034533421676
// MI455X (gfx1250) — compile-verified
//
#include <hip/hip_runtime.h>
#include <cstdint>
#include <cstddef>

#define BLK_CO  32      // output channels per block
#define CI_TILE 16      // input channels staged per LDS chunk
#define BN_EPS  1e-5f

// ---- CDNA5 async global->LDS copy path (ASYNCcnt) -------------------------
__device__ __forceinline__ void async_b32_to_lds(uint32_t lds_byte_addr,
                                                 uint32_t glb_byte_off,
                                                 uint64_t sbase) {
  // GVS mode: mem addr = SADDR(64) + VADDR(32) ; VDST = LDS byte address
  asm volatile("global_load_async_to_lds_b32 %0, %1, %2"
               :: "v"(lds_byte_addr), "v"(glb_byte_off), "s"(sbase)
               : "memory");
}
__device__ __forceinline__ void wait_async_lds() {
  asm volatile("s_wait_asynccnt 0" ::: "memory");
}

// ---------------------------------------------------------------------------
// out[n,co,h,w] = relu( bn( -sum_{ci,kh,kw} |x_pad - w| ) )   (+ 2x2 maxpool)
// Block: 256 threads = 16 quads (2x2 px) x 16 groups; each thread: 2 output
// channels x 4 conv pixels = 8 accumulators. Tile: 8x8 conv outputs x 32 co.
// Double-buffered LDS: chunk k+1's async DMA overlaps chunk k's SAD compute;
// s_wait_asynccnt 0 + barrier drains the prefetch at iteration end.
// ---------------------------------------------------------------------------
template<int CIN, int COUT, int H, int W, bool POOL>
__global__ __launch_bounds__(256)
void adder_bn_relu_layer(const float* __restrict__ x,
                         const float* __restrict__ w,
                         const float* __restrict__ gamma,
                         const float* __restrict__ beta,
                         const float* __restrict__ mean,
                         const float* __restrict__ var,
                         float* __restrict__ out) {
  constexpr int CC     = (CIN < CI_TILE) ? CIN : CI_TILE;  // per-chunk ci count
  constexpr int CHUNKS = (CIN + CI_TILE - 1) / CI_TILE;
  constexpr int TX     = W >> 3;                           // tiles along x (pow2)

  __shared__ float lds_x[2][CI_TILE * 100];        // 10x10 zero-padded tile/ci
  __shared__ float lds_w[2][BLK_CO * CI_TILE * 9]; // [co][ci][tap]

  const int t       = threadIdx.x;
  const int gx0     = (blockIdx.x % TX) << 3;      // TX is pow2 -> bit ops
  const int gy0     = (blockIdx.x / TX) << 3;
  const int co_base = blockIdx.y * BLK_CO;
  const int n       = blockIdx.z;

  // quad mapping (same for POOL and !POOL)
  const int q  = t & 15;                 // 2x2 quad id in the 8x8 tile
  const int g  = t >> 4;                 // channel group (2 co per thread)
  const int qx = (q & 3) << 1;
  const int qy = (q >> 2) << 1;

  // weight-staging mapping: one co row per lane-slice, strided taps (no div)
  const int wj  = t & 31;                // co row 0..31
  const int wr0 = t >> 5;                // first (ci,tap) element, step 8

  float acc[2][4] = {{0.f,0.f,0.f,0.f},{0.f,0.f,0.f,0.f}};

  const uint64_t xbase = (uint64_t)(uintptr_t)(x + (size_t)n * CIN * H * W);
  const uint64_t wbase = (uint64_t)(uintptr_t)w;

  // ---- stage one ci-chunk into buffer b: async DMA in-bounds, ds-zero ring --
  auto stage = [&](int ci0, float* xbuf, float* wbuf) {
    constexpr int XELEM = CC * 100;
    for (int e = t; e < XELEM; e += 256) {
      const int c  = e / 100;            // constant divisors -> mul/shift
      const int r  = e - c * 100;
      const int iy = r / 10;
      const int ix = r - iy * 10;
      const int gy = gy0 + iy - 1;
      const int gx = gx0 + ix - 1;
      const uint32_t laddr = (uint32_t)(uintptr_t)&xbuf[c * 100 + r];
      if ((unsigned)gy < (unsigned)H && (unsigned)gx < (unsigned)W) {
        const uint32_t goff = (uint32_t)((((ci0 + c) * H + gy) * W + gx) * 4);
        async_b32_to_lds(laddr, goff, xbase);
      } else {
        xbuf[c * 100 + r] = 0.f;         // padding ring: disjoint from DMA
      }
    }
    constexpr int WELEM = CC * 9;        // contiguous (ci,tap) run per co
    const uint32_t wrow_goff  = (uint32_t)((((co_base + wj) * CIN + ci0) * 9) * 4);
    const uint32_t wrow_lbase = (uint32_t)(uintptr_t)&wbuf[wj * (CI_TILE * 9)];
    for (int r = wr0; r < WELEM; r += 8)
      async_b32_to_lds(wrow_lbase + (uint32_t)(r * 4),
                       wrow_goff  + (uint32_t)(r * 4), wbase);
  };

  // ------------------------- SAD accumulation on one buffer -----------------
  auto compute = [&](const float* xbuf, const float* wbuf) {
    for (int c = 0; c < CC; ++c) {
      float xv[16];                                  // 4x4 window for the quad
      const float* xs = &xbuf[c * 100 + qy * 10 + qx];
      #pragma unroll
      for (int iy = 0; iy < 4; ++iy)
        #pragma unroll
        for (int ix = 0; ix < 4; ++ix)
          xv[iy * 4 + ix] = xs[iy * 10 + ix];

      #pragma unroll
      for (int j = 0; j < 2; ++j) {
        float wrg[9];
        const float* ws = &wbuf[(g * 2 + j) * (CI_TILE * 9) + c * 9];
        #pragma unroll
        for (int tap = 0; tap < 9; ++tap) wrg[tap] = ws[tap];
        #pragma unroll
        for (int dy = 0; dy < 2; ++dy)
          #pragma unroll
          for (int dx = 0; dx < 2; ++dx) {
            float a = acc[j][dy * 2 + dx];
            #pragma unroll
            for (int kh = 0; kh < 3; ++kh)
              #pragma unroll
              for (int kw = 0; kw < 3; ++kw)
                a -= fabsf(xv[(dy + kh) * 4 + (dx + kw)] - wrg[kh * 3 + kw]);
            acc[j][dy * 2 + dx] = a;
          }
      }
    }
  };

  // ---- software-pipelined chunk loop: DMA(k+1) overlaps compute(k) ----------
  stage(0, lds_x[0], lds_w[0]);
  wait_async_lds();
  __syncthreads();
  for (int ch = 0; ch < CHUNKS; ++ch) {
    const int nb = (ch + 1) & 1;
    if (ch + 1 < CHUNKS)
      stage((ch + 1) * CI_TILE, lds_x[nb], lds_w[nb]);   // prefetch next chunk
    compute(lds_x[ch & 1], lds_w[ch & 1]);               // overlap with DMA
    if (ch + 1 < CHUNKS) {
      wait_async_lds();                                  // drain prefetch only
      __syncthreads();                                   // buffer rotation fence
    }
  }

  // ---------------------- BN + ReLU (+ maxpool) epilogue --------------------
  #pragma unroll
  for (int j = 0; j < 2; ++j) {
    const int co  = co_base + g * 2 + j;
    const float inv = gamma[co] * rsqrtf(var[co] + BN_EPS);
    const float sh  = beta[co] - mean[co] * inv;
    if (!POOL) {
      #pragma unroll
      for (int dy = 0; dy < 2; ++dy)
        #pragma unroll
        for (int dx = 0; dx < 2; ++dx) {
          const float y = fmaxf(fmaf(acc[j][dy * 2 + dx], inv, sh), 0.f);
          const int oy = gy0 + qy + dy, ox = gx0 + qx + dx;
          out[(((size_t)n * COUT + co) * H + oy) * W + ox] = y;
        }
    } else {
      float ymax = -3.402823466e+38f;
      #pragma unroll
      for (int p = 0; p < 4; ++p)
        ymax = fmaxf(ymax, fmaf(acc[j][p], inv, sh));
      const float yv = fmaxf(ymax, 0.f);             // max(relu) == relu(max)
      const int oy = (gy0 >> 1) + (q >> 2), ox = (gx0 >> 1) + (q & 3);
      out[(((size_t)n * COUT + co) * (H >> 1) + oy) * (W >> 1) + ox] = yv;
    }
  }
}

// ---------------------------------------------------------------------------
template<int CIN, int COUT, int H, int W, bool POOL>
static void launch_layer(const float* src, void* const* d_in, int i,
                         float* dst, hipStream_t stream) {
  const float* w     = (const float*)d_in[1 + 5 * i];
  const float* gamma = (const float*)d_in[2 + 5 * i];
  const float* beta  = (const float*)d_in[3 + 5 * i];
  const float* mean  = (const float*)d_in[4 + 5 * i];
  const float* var   = (const float*)d_in[5 + 5 * i];
  dim3 grid((H / 8) * (W / 8), COUT / BLK_CO, 64);
  adder_bn_relu_layer<CIN, COUT, H, W, POOL><<<grid, dim3(256), 0, stream>>>(
      src, w, gamma, beta, mean, var, dst);
}

extern "C" void kernel_launch(void* const* d_in, const int* in_sizes, int n_in,
                              void* d_out, int out_size, void* d_ws, size_t ws_size,
                              hipStream_t stream) {
  (void)in_sizes; (void)n_in; (void)out_size; (void)ws_size;

  const float* x = (const float*)d_in[0];            // (64,3,32,32)
  char*  ws   = (char*)d_ws;
  float* bufA = (float*)ws;                          // up to 8 MB (even layers)
  float* bufB = (float*)(ws + (size_t)(8u << 20));   // up to 2 MB (odd layers)
  float* outp = (float*)d_out;

  launch_layer<  3,  32, 32, 32, false>(x,    d_in, 0, bufA, stream);
  launch_layer< 32,  32, 32, 32, true >(bufA, d_in, 1, bufB, stream);
  launch_layer< 32,  64, 16, 16, false>(bufB, d_in, 2, bufA, stream);
  launch_layer< 64,  64, 16, 16, true >(bufA, d_in, 3, bufB, stream);
  launch_layer< 64, 128,  8,  8, false>(bufB, d_in, 4, bufA, stream);
  launch_layer<128, 128,  8,  8, false>(bufA, d_in, 5, bufB, stream);
  launch_layer<128, 128,  8,  8, false>(bufB, d_in, 6, bufA, stream);
  launch_layer<128, 128,  8,  8, true >(bufA, d_in, 7, outp, stream);
}